// PointerGeneratorDecoder_48782238548562
// MI455X (gfx1250) — compile-verified
//
#include <hip/hip_runtime.h>
#include <hip/hip_bf16.h>
#include <math.h>
#include <stdint.h>

// Problem constants (match reference)
#define H2v  512      // internal hidden
#define Ev   128      // embedding size
#define Vv   50000    // vocab
#define Lv   512      // encoder length
#define Bv   128      // batch
#define VXv  (Vv + Lv) // extended vocab = 50512

typedef __attribute__((ext_vector_type(2))) float v2f;
typedef __attribute__((ext_vector_type(8))) float v8f;

// ---------------------------------------------------------------------------
// Wave-level fp32 WMMA tile-GEMM helper:  acc(16x16) += A[mBase:,:K] * W^T
// where W rows are the output columns (PyTorch x @ W.T convention).
// Layouts per CDNA5 ISA 7.12.2:
//   A 16x4 fp32: lane&15 = M, lane>>4 selects K pair {0,1}|{2,3}
//   B 4x16 fp32: lane&15 = N, lane>>4 selects K pair
//   C/D 16x16 fp32: lane&15 = N, VGPR r -> M = r + 8*(lane>>4)
// ---------------------------------------------------------------------------
__device__ inline void gemm_wT(const float* __restrict__ A, int lda,
                               const float* __restrict__ Wr, int ldw,
                               int mBase, int K, int lane, v8f& acc)
{
    const int half = lane >> 4;
    const float* Arow = A + (size_t)(mBase + (lane & 15)) * lda + half * 2;
    const float* Wrow = Wr + (size_t)(lane & 15) * ldw + half * 2;
    for (int k = 0; k < K; k += 4) {
        v2f a; a.x = Arow[k]; a.y = Arow[k + 1];
        v2f b; b.x = Wrow[k]; b.y = Wrow[k + 1];
        acc = __builtin_amdgcn_wmma_f32_16x16x4_f32(
            /*neg_a=*/false, a, /*neg_b=*/false, b,
            /*c_mod=*/(short)0, acc, /*reuse_a=*/false, /*reuse_b=*/false);
    }
}

// Same, but the B tile lives in LDS (row stride LDSB_STRIDE floats, padded so
// 16 lanes x b64 hit 64 distinct banks: 132 mod 64 = 4 -> banks 4*row+k, all
// distinct across the half-wave, union of both halves covers 64 banks).
#define LDSB_STRIDE 132
__device__ inline void gemm_ldsB(const float* __restrict__ A, int lda,
                                 const float* Bl,
                                 int mBase, int K, int lane, v8f& acc)
{
    const int half = lane >> 4;
    const float* Arow = A + (size_t)(mBase + (lane & 15)) * lda + half * 2;
    const float* Brow = Bl + (lane & 15) * LDSB_STRIDE + half * 2;
    for (int k = 0; k < K; k += 4) {
        v2f a; a.x = Arow[k]; a.y = Arow[k + 1];
        v2f b; b.x = Brow[k]; b.y = Brow[k + 1];
        acc = __builtin_amdgcn_wmma_f32_16x16x4_f32(
            false, a, false, b, (short)0, acc, false, false);
    }
}

// ---------------------------------------------------------------------------
// 0) embedding gather: embW[b,e] = embedding[input_ids[b], e]
// ---------------------------------------------------------------------------
__global__ void gather_kernel(const int* __restrict__ ids,
                              const float* __restrict__ embedding,
                              float* __restrict__ embW)
{
    int idx = blockIdx.x * 256 + threadIdx.x;
    if (idx >= Bv * Ev) return;
    int b = idx >> 7, e = idx & (Ev - 1);
    embW[idx] = embedding[(size_t)ids[b] * Ev + e];
}

// ---------------------------------------------------------------------------
// 1) GRU step (gate order r,z,n).  grid.x = 32 (H2 col tiles), 8 waves = row tiles.
//    6 tile-GEMMs accumulated into 4 v8f accumulators via fp32 WMMA.
// ---------------------------------------------------------------------------
__global__ __launch_bounds__(256) void gru_kernel(
    const float* __restrict__ emb, const float* __restrict__ h0,
    const float* __restrict__ W_ih, const float* __restrict__ W_hh,
    const float* __restrict__ b_ih, const float* __restrict__ b_hh,
    float* __restrict__ hnew_ws, float* __restrict__ hnew_out)
{
    const int lane = threadIdx.x & 31;
    const int wave = threadIdx.x >> 5;     // row tile 0..7
    const int jt   = blockIdx.x;           // col tile 0..31
    const int half = lane >> 4, ln15 = lane & 15;
    const int mBase = wave * 16, j0 = jt * 16;

    v8f cr = {}, cz = {}, cnx = {}, cnh = {};
    // x-part: emb[B,E] @ W_ih[3H2,E]^T (three gate slabs)
    gemm_wT(emb, Ev, W_ih + (size_t)(0 * H2v + j0) * Ev, Ev, mBase, Ev, lane, cr);
    gemm_wT(emb, Ev, W_ih + (size_t)(1 * H2v + j0) * Ev, Ev, mBase, Ev, lane, cz);
    gemm_wT(emb, Ev, W_ih + (size_t)(2 * H2v + j0) * Ev, Ev, mBase, Ev, lane, cnx);
    // h-part: h0[B,H2] @ W_hh[3H2,H2]^T
    gemm_wT(h0, H2v, W_hh + (size_t)(0 * H2v + j0) * H2v, H2v, mBase, H2v, lane, cr);
    gemm_wT(h0, H2v, W_hh + (size_t)(1 * H2v + j0) * H2v, H2v, mBase, H2v, lane, cz);
    gemm_wT(h0, H2v, W_hh + (size_t)(2 * H2v + j0) * H2v, H2v, mBase, H2v, lane, cnh);

#pragma unroll
    for (int r = 0; r < 8; ++r) {
        int mm  = mBase + r + half * 8;
        int col = j0 + ln15;
        float xr = cr[r]  + b_ih[col]            + b_hh[col];
        float xz = cz[r]  + b_ih[H2v + col]      + b_hh[H2v + col];
        float xn = cnx[r] + b_ih[2 * H2v + col];
        float hn = cnh[r] + b_hh[2 * H2v + col];
        float rg = 1.f / (1.f + expf(-xr));
        float zg = 1.f / (1.f + expf(-xz));
        float ng = tanhf(xn + rg * hn);
        float hp = h0[(size_t)mm * H2v + col];
        float hv = (1.f - zg) * ng + zg * hp;
        hnew_ws[(size_t)mm * H2v + col]  = hv;
        hnew_out[(size_t)mm * H2v + col] = hv;   // output #2
    }
}

// ---------------------------------------------------------------------------
// 2) dsl = h_new @ W_ds^T + b_ds   [128,512]
// ---------------------------------------------------------------------------
__global__ __launch_bounds__(256) void dsl_kernel(
    const float* __restrict__ hnew, const float* __restrict__ W_ds,
    const float* __restrict__ b_ds, float* __restrict__ dsl)
{
    const int lane = threadIdx.x & 31, wave = threadIdx.x >> 5;
    const int j0 = blockIdx.x * 16, mBase = wave * 16;
    const int half = lane >> 4, ln15 = lane & 15;
    v8f c = {};
    gemm_wT(hnew, H2v, W_ds + (size_t)j0 * H2v, H2v, mBase, H2v, lane, c);
#pragma unroll
    for (int r = 0; r < 8; ++r) {
        int mm = mBase + r + half * 8, col = j0 + ln15;
        dsl[(size_t)mm * H2v + col] = c[r] + b_ds[col];
    }
}

// ---------------------------------------------------------------------------
// 3) scores[l,b] = sum_h att_v[h]*tanh(w_h[h]*enc[l,b,h] + dsl[b,h])
//    one wave per (l,b); 134MB streaming read -> memory bound.
// ---------------------------------------------------------------------------
__global__ __launch_bounds__(256) void scores_kernel(
    const float* __restrict__ enc, const float* __restrict__ dsl,
    const float* __restrict__ w_h, const float* __restrict__ att_v,
    float* __restrict__ scores)
{
    const int lane = threadIdx.x & 31;
    const int w = blockIdx.x * 8 + (threadIdx.x >> 5);  // 0..65535
    const int l = w >> 7, b = w & (Bv - 1);
    const float* e = enc + (size_t)(l * Bv + b) * H2v;
    const float* d = dsl + (size_t)b * H2v;
    float acc = 0.f;
    for (int i = 0; i < H2v; i += 32) {
        int h = i + lane;
        acc += att_v[h] * tanhf(w_h[h] * e[h] + d[h]);
    }
#pragma unroll
    for (int off = 16; off; off >>= 1) acc += __shfl_xor(acc, off, 32);
    if (lane == 0) scores[l * Bv + b] = acc;
}

// ---------------------------------------------------------------------------
// 4) softmax over BATCH axis (dim=1, faithful to source): per l over 128 b's.
// ---------------------------------------------------------------------------
__global__ __launch_bounds__(128) void softmax_b_kernel(
    const float* __restrict__ scores, float* __restrict__ attn_ws,
    float* __restrict__ attn_out)
{
    __shared__ float red[128];
    const int l = blockIdx.x, t = threadIdx.x;
    float x = scores[l * Bv + t];
    red[t] = x; __syncthreads();
    for (int s = 64; s; s >>= 1) { if (t < s) red[t] = fmaxf(red[t], red[t + s]); __syncthreads(); }
    float mx = red[0]; __syncthreads();
    float e = expf(x - mx);
    red[t] = e; __syncthreads();
    for (int s = 64; s; s >>= 1) { if (t < s) red[t] += red[t + s]; __syncthreads(); }
    float a = e / red[0];
    attn_ws[l * Bv + t]  = a;
    attn_out[l * Bv + t] = a;    // output #3
}

// ---------------------------------------------------------------------------
// 5) context[b,h] = sum_l attn[l,b]*enc[l,b,h]; block per b, thread per h.
//    enc re-read hits the 192MB L2; prefetch 8 rows ahead to hide L2 latency
//    (emits gfx1250 global_prefetch_b8).
// ---------------------------------------------------------------------------
__global__ __launch_bounds__(512) void context_kernel(
    const float* __restrict__ attn, const float* __restrict__ enc,
    float* __restrict__ ctx)
{
    const int b = blockIdx.x, h = threadIdx.x;
    float acc = 0.f;
    for (int l = 0; l < Lv; ++l) {
        if (l + 8 < Lv)
            __builtin_prefetch(&enc[(size_t)((l + 8) * Bv + b) * H2v + h], 0, 0);
        acc += attn[l * Bv + b] * enc[(size_t)(l * Bv + b) * H2v + h];
    }
    ctx[(size_t)b * H2v + h] = acc;
}

// ---------------------------------------------------------------------------
// 6) ohl = [h_new|context] @ W_oh^T + b_oh   [128,128]; K split at 512.
// ---------------------------------------------------------------------------
__global__ __launch_bounds__(256) void oh_kernel(
    const float* __restrict__ hnew, const float* __restrict__ ctx,
    const float* __restrict__ W_oh, const float* __restrict__ b_oh,
    float* __restrict__ ohl)
{
    const int lane = threadIdx.x & 31, wave = threadIdx.x >> 5;
    const int j0 = blockIdx.x * 16, mBase = wave * 16;
    const int half = lane >> 4, ln15 = lane & 15;
    v8f c = {};
    gemm_wT(hnew, H2v, W_oh + (size_t)j0 * (2 * H2v),        2 * H2v, mBase, H2v, lane, c);
    gemm_wT(ctx,  H2v, W_oh + (size_t)j0 * (2 * H2v) + H2v,  2 * H2v, mBase, H2v, lane, c);
#pragma unroll
    for (int r = 0; r < 8; ++r) {
        int mm = mBase + r + half * 8, col = j0 + ln15;
        ohl[(size_t)mm * Ev + col] = c[r] + b_oh[col];
    }
}

// ---------------------------------------------------------------------------
// 6b) p_gen[b] = sigmoid([combined|emb] . W_ptr + b_ptr)
// ---------------------------------------------------------------------------
__global__ __launch_bounds__(128) void pgen_kernel(
    const float* __restrict__ hnew, const float* __restrict__ ctx,
    const float* __restrict__ emb, const float* __restrict__ W_ptr,
    const float* __restrict__ b_ptr, float* __restrict__ pgen)
{
    const int b = threadIdx.x;
    float s = b_ptr[0];
    for (int i = 0; i < H2v; ++i) s += hnew[(size_t)b * H2v + i] * W_ptr[i];
    for (int i = 0; i < H2v; ++i) s += ctx[(size_t)b * H2v + i] * W_ptr[H2v + i];
    for (int i = 0; i < Ev;  ++i) s += emb[(size_t)b * Ev + i] * W_ptr[2 * H2v + i];
    pgen[b] = 1.f / (1.f + expf(-s));
}

// ---------------------------------------------------------------------------
// 7a) vocab logits = ohl @ W_ov^T + b_ov  [128,50000]; the dominant-traffic
//     GEMM (W_ov = 25.6MB streamed once, zero reuse). Stage each 16x128 B-tile
//     (8KB) into LDS with the CDNA5 async global->LDS path:
//       global_load_async_to_lds_b128 (ASYNCcnt) -> s_wait_asynccnt 0 ->
//       workgroup barrier -> WMMA fed by conflict-free ds_load_b64.
//     A (ohl, 64KB) is fully L2-resident and reused by all 3125 blocks, so it
//     stays on direct global loads.
// ---------------------------------------------------------------------------
__global__ __launch_bounds__(256) void vocab_kernel(
    const float* __restrict__ ohl, const float* __restrict__ W_ov,
    const float* __restrict__ b_ov, float* __restrict__ logits)
{
    __shared__ float ldsB[16 * LDSB_STRIDE];   // 16 rows, padded stride

    const int lane = threadIdx.x & 31, wave = threadIdx.x >> 5;
    const int v0 = blockIdx.x * 16, mBase = wave * 16;
    const int half = lane >> 4, ln15 = lane & 15;

    // Cooperative async copy: 16 rows x 128 floats = 512 b128 chunks / 256 thr.
    for (int c = threadIdx.x; c < 512; c += 256) {
        int row  = c >> 5;           // 0..15
        int col4 = (c & 31) * 4;     // 0..124, 16B aligned
        const float* g = W_ov + (size_t)(v0 + row) * Ev + col4;
        unsigned lds_off = (unsigned)(uintptr_t)&ldsB[row * LDSB_STRIDE + col4];
        asm volatile("global_load_async_to_lds_b128 %0, %1, off"
                     :: "v"(lds_off), "v"((unsigned long long)(uintptr_t)g)
                     : "memory");
    }
    asm volatile("s_wait_asynccnt 0x0" ::: "memory");
    __syncthreads();

    v8f c = {};
    gemm_ldsB(ohl, Ev, ldsB, mBase, Ev, lane, c);
#pragma unroll
    for (int r = 0; r < 8; ++r) {
        int mm = mBase + r + half * 8, vv = v0 + ln15;
        logits[(size_t)mm * Vv + vv] = c[r] + b_ov[vv];
    }
}

// ---------------------------------------------------------------------------
// 7b) per-row max & sum(exp) over 50000 vocab logits
// ---------------------------------------------------------------------------
__global__ __launch_bounds__(256) void rowred_kernel(
    const float* __restrict__ logits, float* __restrict__ rmax,
    float* __restrict__ rsum)
{
    __shared__ float red[256];
    const int b = blockIdx.x, t = threadIdx.x;
    float mx = -3.402823466e38f;
    for (int v = t; v < Vv; v += 256) mx = fmaxf(mx, logits[(size_t)b * Vv + v]);
    red[t] = mx; __syncthreads();
    for (int s = 128; s; s >>= 1) { if (t < s) red[t] = fmaxf(red[t], red[t + s]); __syncthreads(); }
    mx = red[0]; __syncthreads();
    float sm = 0.f;
    for (int v = t; v < Vv; v += 256) sm += expf(logits[(size_t)b * Vv + v] - mx);
    red[t] = sm; __syncthreads();
    for (int s = 128; s; s >>= 1) { if (t < s) red[t] += red[t + s]; __syncthreads(); }
    if (t == 0) { rmax[b] = mx; rsum[b] = red[0]; }
}

// ---------------------------------------------------------------------------
// 7c) p_final gen part: p_vocab*p_gen for v<V, zeros for pad region
// ---------------------------------------------------------------------------
__global__ __launch_bounds__(256) void final_kernel(
    const float* __restrict__ logits, const float* __restrict__ rmax,
    const float* __restrict__ rsum, const float* __restrict__ pgen,
    float* __restrict__ out)
{
    const int v = blockIdx.x * 256 + threadIdx.x;
    const int b = blockIdx.y;
    if (v >= VXv) return;
    float val = 0.f;
    if (v < Vv)
        val = expf(logits[(size_t)b * Vv + v] - rmax[b]) * (pgen[b] / rsum[b]);
    out[(size_t)b * VXv + v] = val;
}

// ---------------------------------------------------------------------------
// 7d) copy distribution scatter (deterministic: one thread per b, serial in l)
// ---------------------------------------------------------------------------
__global__ __launch_bounds__(128) void scatter_kernel(
    const int* __restrict__ full_input, const float* __restrict__ attn,
    const float* __restrict__ pgen, float* __restrict__ out)
{
    const int b = threadIdx.x;
    const float g = 1.f - pgen[b];
    for (int l = 0; l < Lv; ++l) {
        int c = full_input[l * Bv + b];
        out[(size_t)b * VXv + c] += g * attn[l * Bv + b];
    }
}

// ---------------------------------------------------------------------------
// host-side launch
// ---------------------------------------------------------------------------
extern "C" void kernel_launch(void* const* d_in, const int* in_sizes, int n_in,
                              void* d_out, int out_size, void* d_ws, size_t ws_size,
                              hipStream_t stream)
{
    (void)in_sizes; (void)n_in; (void)out_size; (void)ws_size;
    const int*   input_ids = (const int*)  d_in[0];
    const float* hidden    = (const float*)d_in[1];   // [1,B,H2]
    const float* enc       = (const float*)d_in[2];   // [L,B,H2]
    const int*   full_inp  = (const int*)  d_in[3];   // [L,B]
    const float* embedding = (const float*)d_in[4];
    const float* W_ih      = (const float*)d_in[5];
    const float* W_hh      = (const float*)d_in[6];
    const float* b_ih      = (const float*)d_in[7];
    const float* b_hh      = (const float*)d_in[8];
    const float* W_ds      = (const float*)d_in[9];
    const float* b_ds      = (const float*)d_in[10];
    const float* w_h       = (const float*)d_in[11];
    const float* att_v     = (const float*)d_in[12];
    const float* W_oh      = (const float*)d_in[13];
    const float* b_oh      = (const float*)d_in[14];
    const float* W_ov      = (const float*)d_in[15];
    const float* b_ov      = (const float*)d_in[16];
    const float* W_ptr     = (const float*)d_in[17];
    const float* b_ptr     = (const float*)d_in[18];

    float* out = (float*)d_out;
    float* ws  = (float*)d_ws;

    // d_out layout: p_final [128,50512] | h_new [1,128,512] | attn [512,128]
    float* out_p    = out;
    float* out_h    = out + (size_t)Bv * VXv;
    float* out_attn = out_h + (size_t)Bv * H2v;

    // workspace layout (floats)
    float* emb_ws   = ws;                       // 16384
    float* hnew_ws  = emb_ws  + 16384;          // 65536
    float* dsl_ws   = hnew_ws + 65536;          // 65536
    float* sc_ws    = dsl_ws  + 65536;          // 65536
    float* attn_ws  = sc_ws   + 65536;          // 65536
    float* ctx_ws   = attn_ws + 65536;          // 65536
    float* ohl_ws   = ctx_ws  + 65536;          // 16384
    float* pgen_ws  = ohl_ws  + 16384;          // 128
    float* rmax_ws  = pgen_ws + 128;            // 128
    float* rsum_ws  = rmax_ws + 128;            // 128
    float* logit_ws = rsum_ws + 128;            // 6,400,000 (~27MB total)

    gather_kernel   <<<64, 256, 0, stream>>>(input_ids, embedding, emb_ws);
    gru_kernel      <<<32, 256, 0, stream>>>(emb_ws, hidden, W_ih, W_hh, b_ih, b_hh,
                                             hnew_ws, out_h);
    dsl_kernel      <<<32, 256, 0, stream>>>(hnew_ws, W_ds, b_ds, dsl_ws);
    scores_kernel   <<<8192, 256, 0, stream>>>(enc, dsl_ws, w_h, att_v, sc_ws);
    softmax_b_kernel<<<512, 128, 0, stream>>>(sc_ws, attn_ws, out_attn);
    context_kernel  <<<128, 512, 0, stream>>>(attn_ws, enc, ctx_ws);
    oh_kernel       <<<8, 256, 0, stream>>>(hnew_ws, ctx_ws, W_oh, b_oh, ohl_ws);
    pgen_kernel     <<<1, 128, 0, stream>>>(hnew_ws, ctx_ws, emb_ws, W_ptr, b_ptr, pgen_ws);
    vocab_kernel    <<<3125, 256, 0, stream>>>(ohl_ws, W_ov, b_ov, logit_ws);
    rowred_kernel   <<<128, 256, 0, stream>>>(logit_ws, rmax_ws, rsum_ws);
    final_kernel    <<<dim3((VXv + 255) / 256, Bv), 256, 0, stream>>>(
                        logit_ws, rmax_ws, rsum_ws, pgen_ws, out_p);
    scatter_kernel  <<<1, 128, 0, stream>>>(full_inp, attn_ws, pgen_ws, out_p);
}